// DistilBertCRF_57509612093731
// MI455X (gfx1250) — compile-verified
//
#include <hip/hip_runtime.h>

// ---------------------------------------------------------------------------
// DistilBERT classifier head + CRF (loss + viterbi decode) for MI455X/gfx1250
//
//  Kernel 1: logits = hidden @ W^T + b  via v_wmma_f32_16x16x4_f32 (wave32)
//            -> HBM-bound (100 MB of hidden_state @ 23.3 TB/s ~= 4.3 us).
//            One global_load_b128 per lane feeds TWO WMMAs (K-slot
//            permutation is legal since A and B share the same lane map).
//  Kernel 2: per-batch CRF: numerator, logsumexp forward scan, viterbi scan
//            + LDS backtrace.  One wave per batch, 16 lanes = 16 CRF states.
//  Kernel 3: deterministic loss reduction (no fp atomics).
// ---------------------------------------------------------------------------

typedef __attribute__((ext_vector_type(2))) float v2f;
typedef __attribute__((ext_vector_type(4))) float v4f;
typedef __attribute__((ext_vector_type(8))) float v8f;

#define Bsz  64
#define Lseq 512
#define Hdim 768
#define Ttag 16
#define Mrows (Bsz * Lseq)          // 32768 logit rows

// ---------------------------------------------------------------------------
// Kernel 1: WMMA GEMM.  Each wave owns a 16(M)x16(N) output tile; K=768 in
// steps of 8 -> 96 iterations x 2 v_wmma_f32_16x16x4_f32.
//
// f32 A fragment layout (ISA 7.12.2, 16x4 MxK, 2 VGPRs):
//   lane = kg*16 + m; VGPR0 = elem(m, 2*kg), VGPR1 = elem(m, 2*kg+1).
// We load float4 at column (k + 4*kg) per lane; (f.x,f.y) feeds WMMA#0 and
// (f.z,f.w) feeds WMMA#1.  That permutes which physical column each K-slot
// carries, but B is built from W[n, k + 4*kg] with the identical lane
// striping, so the permutation cancels in the K-contraction.
// ---------------------------------------------------------------------------
__global__ __launch_bounds__(256) void gemm_logits_wmma(
    const float* __restrict__ hidden,   // [Mrows, Hdim]
    const float* __restrict__ W,        // [Ttag,  Hdim]
    const float* __restrict__ bias,     // [Ttag]
    float* __restrict__ logits)         // [Mrows, Ttag]
{
  const int lane = threadIdx.x & 31;
  const int wave = threadIdx.x >> 5;
  const int tile = blockIdx.x * 8 + wave;     // 2048 tiles total
  const int rb   = tile * 16;                 // first output row of tile
  const int m    = lane & 15;                 // M row (A) / N col (B/D)
  const int kg   = lane >> 4;                 // K group within fragment

  const float* __restrict__ arow = hidden + (size_t)(rb + m) * Hdim + 4 * kg;
  const float* __restrict__ brow = W      + (size_t)m        * Hdim + 4 * kg;

  v8f c = {0.f, 0.f, 0.f, 0.f, 0.f, 0.f, 0.f, 0.f};

#pragma unroll 4
  for (int k = 0; k < Hdim; k += 8) {
    v4f a = *(const v4f*)(arow + k);          // global_load_b128 (aligned 16B)
    v4f b = *(const v4f*)(brow + k);          // global_load_b128 (L2-resident)
    v2f a0 = {a.x, a.y}, a1 = {a.z, a.w};
    v2f b0 = {b.x, b.y}, b1 = {b.z, b.w};
    // (neg_a, A, neg_b, B, c_mod, C, reuse_a, reuse_b)
    c = __builtin_amdgcn_wmma_f32_16x16x4_f32(
        false, a0, false, b0, (short)0, c, false, false);
    c = __builtin_amdgcn_wmma_f32_16x16x4_f32(
        false, a1, false, b1, (short)0, c, false, false);
  }

  // D layout: VGPR i -> M = i (lanes 0-15) / i+8 (lanes 16-31); N = lane&15.
  const float bv = bias[m];
#pragma unroll
  for (int i = 0; i < 8; ++i) {
    const int row = rb + i + 8 * kg;
    logits[(size_t)row * Ttag + m] = c[i] + bv;
  }
}

// ---------------------------------------------------------------------------
// Kernel 2: fused CRF per batch element.  Block = 1 wave (32 lanes); lanes
// 0-15 carry the 16 CRF states (lanes 16-31 mirror them, keeping EXEC full).
// trans column j lives in 16 VGPRs per lane; previous score vector is
// broadcast with wave shuffles.  Viterbi history in LDS for fast backtrace.
// ---------------------------------------------------------------------------
__global__ __launch_bounds__(32) void crf_fused(
    const float* __restrict__ logits,      // [Bsz*Lseq, Ttag]
    const int*   __restrict__ mask,        // [Bsz, Lseq]
    const int*   __restrict__ labels,      // [Bsz, Lseq]
    const float* __restrict__ start_trans, // [Ttag]
    const float* __restrict__ end_trans,   // [Ttag]
    const float* __restrict__ trans,       // [Ttag, Ttag]
    float* __restrict__ llh,               // [Bsz]
    float* __restrict__ tags_out)          // [Bsz, Lseq] (float)
{
  __shared__ unsigned char hist[Lseq][Ttag];   // 8 KB viterbi backpointers

  const int b    = blockIdx.x;
  const int lane = threadIdx.x;
  const int j    = lane & 15;                  // this lane's CRF state

  const float* __restrict__ em   = logits + (size_t)b * Lseq * Ttag;
  const int*   __restrict__ mrow = mask   + b * Lseq;
  const int*   __restrict__ lrow = labels + b * Lseq;

  float trc[16];                               // trans[:, j]
#pragma unroll
  for (int i = 0; i < 16; ++i) trc[i] = trans[i * 16 + j];

  float score  = start_trans[j] + em[j];       // forward logsumexp state
  float vscore = score;                        // viterbi state

  int   lab_prev = lrow[0];
  float numer    = start_trans[lab_prev] + em[lab_prev];
  int   msum     = mrow[0];

  for (int t = 1; t < Lseq; ++t) {
    const int   mi = mrow[t];
    const float mf = (float)mi;
    msum += mi;
    const float e = em[t * Ttag + j];

    // ---- forward: score'[j] = logsumexp_i(score[i] + trans[i][j]) + em ----
    float vals[16];
    float mx = -3.4e38f;
#pragma unroll
    for (int i = 0; i < 16; ++i) {
      const float si = __shfl(score, i, 32);   // broadcast state i
      vals[i] = si + trc[i];
      mx = fmaxf(mx, vals[i]);
    }
    float sum = 0.f;
#pragma unroll
    for (int i = 0; i < 16; ++i) sum += __expf(vals[i] - mx);
    const float nxt = mx + __logf(sum) + e;
    score = (mi > 0) ? nxt : score;

    // ---- viterbi: argmax_i(vscore[i] + trans[i][j]) (first-max wins) ----
    float bv = -3.4e38f;
    int   ba = 0;
#pragma unroll
    for (int i = 0; i < 16; ++i) {
      const float vv = __shfl(vscore, i, 32) + trc[i];
      if (vv > bv) { bv = vv; ba = i; }
    }
    if (lane < 16) hist[t][j] = (unsigned char)ba;
    vscore = bv + e;

    // ---- numerator (uniform across lanes -> scalarizable loads) ----
    const int lab = lrow[t];
    numer += mf * (trans[lab_prev * 16 + lab] + em[t * Ttag + lab]);
    lab_prev = lab;
  }

  const int last_tag = lrow[msum - 1];
  numer += end_trans[last_tag];

  // ---- denominator: logsumexp over final scores + end_trans ----
  const float f  = score + end_trans[j];
  float fm = -3.4e38f;
#pragma unroll
  for (int i = 0; i < 16; ++i) fm = fmaxf(fm, __shfl(f, i, 32));
  float fs = 0.f;
#pragma unroll
  for (int i = 0; i < 16; ++i) fs += __expf(__shfl(f, i, 32) - fm);
  const float denom = fm + __logf(fs);

  // ---- viterbi terminal argmax ----
  const float ff = vscore + end_trans[j];
  float bestv = -3.4e38f;
  int   bestj = 0;
#pragma unroll
  for (int i = 0; i < 16; ++i) {
    const float v = __shfl(ff, i, 32);
    if (v > bestv) { bestv = v; bestj = i; }
  }

  __syncthreads();                         // hist visible for backtrace

  if (lane == 0) {
    llh[b] = numer - denom;
    float* __restrict__ trow = tags_out + (size_t)b * Lseq;
    int tag = bestj;
    trow[Lseq - 1] = (float)tag;
    for (int t = Lseq - 1; t >= 1; --t) {  // LDS-latency backtrace
      tag = hist[t][tag];
      trow[t - 1] = (float)tag;
    }
  }
}

// ---------------------------------------------------------------------------
// Kernel 3: deterministic loss = -mean(llh).  64 values -> single lane.
// ---------------------------------------------------------------------------
__global__ __launch_bounds__(32) void loss_reduce(
    const float* __restrict__ llh, float* __restrict__ out)
{
  if (threadIdx.x == 0) {
    float s = 0.f;
    for (int i = 0; i < Bsz; ++i) s += llh[i];
    out[0] = -s / (float)Bsz;
  }
}

// ---------------------------------------------------------------------------
extern "C" void kernel_launch(void* const* d_in, const int* in_sizes, int n_in,
                              void* d_out, int out_size, void* d_ws, size_t ws_size,
                              hipStream_t stream)
{
  const float* hidden = (const float*)d_in[0];  // [64,512,768] f32
  const int*   mask   = (const int*)  d_in[1];  // [64,512] i32
  const int*   labels = (const int*)  d_in[2];  // [64,512] i32
  const float* W      = (const float*)d_in[3];  // [16,768] f32
  const float* bias   = (const float*)d_in[4];  // [16] f32
  const float* st     = (const float*)d_in[5];  // [16] f32
  const float* et     = (const float*)d_in[6];  // [16] f32
  const float* tr     = (const float*)d_in[7];  // [16,16] f32

  float* out    = (float*)d_out;                // [0]=loss, [1..]=tags (B*L)
  float* logits = (float*)d_ws;                                  // 2 MB
  float* llh    = (float*)((char*)d_ws +
                           (size_t)Mrows * Ttag * sizeof(float)); // 256 B

  // 2048 tiles / 8 waves per block = 256 blocks
  gemm_logits_wmma<<<Mrows / 16 / 8, 256, 0, stream>>>(hidden, W, bias, logits);
  crf_fused<<<Bsz, 32, 0, stream>>>(logits, mask, labels, st, et, tr,
                                    llh, out + 1);
  loss_reduce<<<1, 32, 0, stream>>>(llh, out);
}